// GCN1dBlock_11751030522221
// MI455X (gfx1250) — compile-verified
//
#include <hip/hip_runtime.h>

// ---------------------------------------------------------------------------
// GCN 1d block: 3x (GCNConv(improved) -> BatchNorm(batch stats) -> ReLU)
// G=32 graphs, L=2048 nodes, E=16384 edges, channels 64->128->128->64, fp32.
// Dense matmuls use V_WMMA_F32_16X16X4_F32 (fp32 matrix cores, wave32).
// ---------------------------------------------------------------------------

typedef float v2f __attribute__((ext_vector_type(2)));
typedef float v8f __attribute__((ext_vector_type(8)));

#define EPS_BN 1e-5f

// ---------------- graph normalization ----------------

__global__ void k_init_deg(float* __restrict__ deg, int L) {
  int i = blockIdx.x * blockDim.x + threadIdx.x;
  if (i < L) deg[i] = 2.0f;   // improved=True: +2.0 self-loop weight in degree
}

__global__ void k_count_deg(const int* __restrict__ dst, float* __restrict__ deg, int E) {
  int e = blockIdx.x * blockDim.x + threadIdx.x;
  if (e < E) atomicAdd(&deg[dst[e]], 1.0f);   // adds of exactly 1.0f -> order independent
}

__global__ void k_norm(const float* __restrict__ deg, float* __restrict__ dis,
                       float* __restrict__ self_w, int L) {
  int i = blockIdx.x * blockDim.x + threadIdx.x;
  if (i < L) {
    float d = deg[i];
    dis[i] = 1.0f / sqrtf(d);
    self_w[i] = 2.0f / d;
  }
}

// exclusive scan of per-node counts (count = deg - 2) -> csr_start. L==2048, 1 block x 256.
__global__ void k_scan(const float* __restrict__ deg, int* __restrict__ start, int L) {
  __shared__ int chunkSum[256];
  const int t = threadIdx.x;
  const int per = L / 256;          // 8
  const int base = t * per;
  int cnt[8], excl[8];
  int run = 0;
#pragma unroll
  for (int i = 0; i < 8; ++i) cnt[i] = (int)(deg[base + i] - 1.5f);  // truncate(count+0.5)
#pragma unroll
  for (int i = 0; i < 8; ++i) { excl[i] = run; run += cnt[i]; }
  chunkSum[t] = run;
  __syncthreads();
  if (t == 0) {
    int running = 0;
    for (int i = 0; i < 256; ++i) { int v = chunkSum[i]; chunkSum[i] = running; running += v; }
    start[L] = running;             // == E
  }
  __syncthreads();
  const int off = chunkSum[t];
#pragma unroll
  for (int i = 0; i < 8; ++i) start[base + i] = off + excl[i];
}

// stable CSR fill: one thread per dst node scans edges in order (wave-uniform loads).
__global__ void k_fill_csr(const int* __restrict__ srcIdx, const int* __restrict__ dstIdx,
                           const float* __restrict__ dis, const int* __restrict__ start,
                           int* __restrict__ csr_src, float* __restrict__ csr_w,
                           int L, int E) {
  int d = blockIdx.x * blockDim.x + threadIdx.x;
  if (d >= L) return;
  int ptr = start[d];
  const float dd = dis[d];
  for (int e = 0; e < E; ++e) {
    if (dstIdx[e] == d) {
      int s = srcIdx[e];
      csr_src[ptr] = s;
      csr_w[ptr] = dis[s] * dd;
      ++ptr;
    }
  }
}

// ---------------- layout: x[g,c,l] -> h[g,l,c] ----------------

__global__ __launch_bounds__(256) void k_transpose_in(const float* __restrict__ x,
                                                      float* __restrict__ h, int C, int L) {
  int tid = blockIdx.x * blockDim.x + threadIdx.x;
  int l = tid % L;
  int rest = tid / L;
  int c = rest % C;
  int g = rest / C;
  h[((size_t)g * L + l) * C + c] = x[((size_t)g * C + c) * L + l];
}

// ---------------- aggregation (gather over CSR), BN+ReLU of input fused ----------------

__global__ __launch_bounds__(256) void k_aggregate(const float* __restrict__ hin,
    float* __restrict__ aggout, const int* __restrict__ rs, const int* __restrict__ cs,
    const float* __restrict__ cw, const float* __restrict__ sw,
    const float* __restrict__ bna, const float* __restrict__ bnb,
    int C4, int L, int useBN) {
  int tid = blockIdx.x * blockDim.x + threadIdx.x;
  int c4 = tid % C4;
  int rest = tid / C4;
  int l = rest % L;
  int g = rest / L;

  const float4* __restrict__ hin4 = (const float4*)hin;
  float4 a4 = make_float4(1.f, 1.f, 1.f, 1.f);
  float4 b4 = make_float4(0.f, 0.f, 0.f, 0.f);
  if (useBN) {
    a4 = ((const float4*)bna)[c4];
    b4 = ((const float4*)bnb)[c4];
  }
  const size_t gbase = (size_t)g * L * C4;

  // self term
  float4 v = hin4[gbase + (size_t)l * C4 + c4];
  if (useBN) {
    v.x = fmaxf(fmaf(a4.x, v.x, b4.x), 0.f);
    v.y = fmaxf(fmaf(a4.y, v.y, b4.y), 0.f);
    v.z = fmaxf(fmaf(a4.z, v.z, b4.z), 0.f);
    v.w = fmaxf(fmaf(a4.w, v.w, b4.w), 0.f);
  }
  const float s = sw[l];
  float4 acc = make_float4(s * v.x, s * v.y, s * v.z, s * v.w);

  const int e0 = rs[l], e1 = rs[l + 1];
  for (int j = e0; j < e1; ++j) {
    const float wj = cw[j];
    float4 u = hin4[gbase + (size_t)cs[j] * C4 + c4];
    if (useBN) {
      u.x = fmaxf(fmaf(a4.x, u.x, b4.x), 0.f);
      u.y = fmaxf(fmaf(a4.y, u.y, b4.y), 0.f);
      u.z = fmaxf(fmaf(a4.z, u.z, b4.z), 0.f);
      u.w = fmaxf(fmaf(a4.w, u.w, b4.w), 0.f);
    }
    acc.x = fmaf(wj, u.x, acc.x);
    acc.y = fmaf(wj, u.y, acc.y);
    acc.z = fmaf(wj, u.z, acc.z);
    acc.w = fmaf(wj, u.w, acc.w);
  }
  ((float4*)aggout)[gbase + (size_t)l * C4 + c4] = acc;
}

// ---------------- dense matmul via fp32 WMMA ----------------
// Out[M,N] = A[M,K] @ W[K,N] + bias.  Block = 256 threads = 8 waves; each wave
// computes one 16x16 tile; block covers 128 rows x 16 cols. K fully unrolled.
template <int K, int N>
__global__ __launch_bounds__(256) void k_gemm_wmma(const float* __restrict__ A,
    const float* __restrict__ W, const float* __restrict__ bias,
    float* __restrict__ Out) {
  const int wave = threadIdx.x >> 5;
  const int lane = threadIdx.x & 31;
  const int mi   = lane & 15;      // row/col within tile
  const int half = lane >> 4;      // 0: lanes 0-15, 1: lanes 16-31
  const int kh   = half * 2;       // K-pair owned by this half (ISA 16x4 fp32 A layout)
  const int m0 = blockIdx.x * 128 + wave * 16;
  const int n0 = blockIdx.y * 16;

  const float* __restrict__ arow = A + (size_t)(m0 + mi) * K + kh;
  const float* __restrict__ wcol = W + n0 + mi;

  v8f acc = {0.f, 0.f, 0.f, 0.f, 0.f, 0.f, 0.f, 0.f};
#pragma unroll
  for (int k = 0; k < K; k += 4) {
    v2f a = *(const v2f*)(arow + k);              // A[m, k+kh], A[m, k+kh+1]
    v2f b;
    b.x = wcol[(k + kh) * N];                     // B[k+kh,   n]
    b.y = wcol[(k + kh + 1) * N];                 // B[k+kh+1, n]
    acc = __builtin_amdgcn_wmma_f32_16x16x4_f32(
        /*neg_a=*/false, a, /*neg_b=*/false, b,
        /*c_mod=*/(short)0, acc, /*reuse_a=*/false, /*reuse_b=*/false);
  }
  const float bv = bias[n0 + mi];
  float* __restrict__ o = Out + (size_t)(m0 + half * 8) * N + n0 + mi;
#pragma unroll
  for (int r = 0; r < 8; ++r) o[(size_t)r * N] = acc[r] + bv;   // D row M = half*8 + r
}

// ---------------- deterministic BatchNorm statistics ----------------

// grid = M/256 blocks; each block reduces 256 rows into per-channel partials.
__global__ __launch_bounds__(256) void k_stats_partial(const float* __restrict__ h,
    float* __restrict__ ps, float* __restrict__ pq, int C) {
  __shared__ float ssum[256];
  __shared__ float ssq[256];
  const int c = threadIdx.x % C;
  const int sub = threadIdx.x / C;
  const int nsub = blockDim.x / C;
  const int r0 = blockIdx.x * 256;
  float s = 0.f, q = 0.f;
  for (int r = sub; r < 256; r += nsub) {
    float v = h[(size_t)(r0 + r) * C + c];
    s += v;
    q += v * v;
  }
  ssum[threadIdx.x] = s;
  ssq[threadIdx.x] = q;
  __syncthreads();
  if (sub == 0) {
    for (int t = 1; t < nsub; ++t) { s += ssum[t * C + c]; q += ssq[t * C + c]; }
    ps[blockIdx.x * C + c] = s;
    pq[blockIdx.x * C + c] = q;
  }
}

// fold mean/var + gamma/beta into per-channel scale (bna) and shift (bnb).
__global__ void k_stats_final(const float* __restrict__ ps, const float* __restrict__ pq,
                              const float* __restrict__ gamma, const float* __restrict__ beta,
                              float* __restrict__ bna, float* __restrict__ bnb,
                              int C, int nblocks, float invM) {
  int c = threadIdx.x;
  if (c >= C) return;
  float s = 0.f, q = 0.f;
  for (int b = 0; b < nblocks; ++b) { s += ps[b * C + c]; q += pq[b * C + c]; }
  float mean = s * invM;
  float var = q * invM - mean * mean;
  float inv = rsqrtf(var + EPS_BN);
  float a = gamma[c] * inv;
  bna[c] = a;
  bnb[c] = beta[c] - mean * a;
}

// ---------------- final BN+ReLU + transpose to [G, C, L] ----------------

__global__ __launch_bounds__(256) void k_output(const float* __restrict__ h,
    const float* __restrict__ bna, const float* __restrict__ bnb,
    float* __restrict__ out, int C, int L) {
  int tid = blockIdx.x * blockDim.x + threadIdx.x;
  int l = tid % L;
  int rest = tid / L;
  int c = rest % C;
  int g = rest / C;
  float v = h[((size_t)g * L + l) * C + c];
  out[((size_t)g * C + c) * L + l] = fmaxf(fmaf(bna[c], v, bnb[c]), 0.f);
}

// ---------------------------------------------------------------------------

extern "C" void kernel_launch(void* const* d_in, const int* in_sizes, int n_in,
                              void* d_out, int out_size, void* d_ws, size_t ws_size,
                              hipStream_t stream) {
  const float* x  = (const float*)d_in[0];
  const int*   ei = (const int*)d_in[1];
  const float* W1 = (const float*)d_in[2];  const float* b1  = (const float*)d_in[3];
  const float* g1 = (const float*)d_in[4];  const float* be1 = (const float*)d_in[5];
  const float* W2 = (const float*)d_in[6];  const float* b2  = (const float*)d_in[7];
  const float* g2 = (const float*)d_in[8];  const float* be2 = (const float*)d_in[9];
  const float* W3 = (const float*)d_in[10]; const float* b3  = (const float*)d_in[11];
  const float* g3 = (const float*)d_in[12]; const float* be3 = (const float*)d_in[13];

  const int L = 2048, E = 16384, G = 32;
  const int C0 = 64, C1 = 128, C2 = 128, C3 = 64;
  const int M = G * L;                       // 65536 rows

  // --- workspace carve (256B aligned bump allocator) ---
  char* w = (char*)d_ws;
  char* wend = w + ws_size;
  auto alloc = [&](size_t bytes) -> void* {
    void* p = (void*)w;
    w += (bytes + 255) & ~(size_t)255;
    return p;
  };
  float* deg      = (float*)alloc((size_t)L * 4);
  float* dis      = (float*)alloc((size_t)L * 4);
  float* self_w   = (float*)alloc((size_t)L * 4);
  int*   csrStart = (int*)  alloc((size_t)(L + 1) * 4);
  int*   csrSrc   = (int*)  alloc((size_t)E * 4);
  float* csrW     = (float*)alloc((size_t)E * 4);
  float* psum     = (float*)alloc((size_t)256 * 128 * 4);
  float* psq      = (float*)alloc((size_t)256 * 128 * 4);
  float* bnA      = (float*)alloc((size_t)128 * 4);
  float* bnB      = (float*)alloc((size_t)128 * 4);
  float* bufA     = (float*)alloc((size_t)M * 128 * 4);   // 32 MB
  float* bufB     = (float*)alloc((size_t)M * 128 * 4);   // 32 MB
  if (w > wend) return;   // workspace too small; bail deterministically

  const dim3 blk(256);
  const int* srcIdx = ei;          // edge_index[0]
  const int* dstIdx = ei + E;      // edge_index[1]

  // --- graph norm + CSR ---
  k_init_deg<<<(L + 255) / 256, blk, 0, stream>>>(deg, L);
  k_count_deg<<<(E + 255) / 256, blk, 0, stream>>>(dstIdx, deg, E);
  k_norm<<<(L + 255) / 256, blk, 0, stream>>>(deg, dis, self_w, L);
  k_scan<<<1, 256, 0, stream>>>(deg, csrStart, L);
  k_fill_csr<<<(L + 255) / 256, blk, 0, stream>>>(srcIdx, dstIdx, dis, csrStart, csrSrc, csrW, L, E);

  // --- h0 = transpose(x): [G,L,C0] in bufA ---
  k_transpose_in<<<(G * C0 * L) / 256, blk, 0, stream>>>(x, bufA, C0, L);

  // --- layer 1: agg(bufA) -> bufB ; bufB @ W1 -> bufA ; stats ---
  k_aggregate<<<(G * L * (C0 / 4)) / 256, blk, 0, stream>>>(
      bufA, bufB, csrStart, csrSrc, csrW, self_w, nullptr, nullptr, C0 / 4, L, 0);
  k_gemm_wmma<64, 128><<<dim3(M / 128, C1 / 16), blk, 0, stream>>>(bufB, W1, b1, bufA);
  k_stats_partial<<<M / 256, blk, 0, stream>>>(bufA, psum, psq, C1);
  k_stats_final<<<1, 128, 0, stream>>>(psum, psq, g1, be1, bnA, bnB, C1, M / 256, 1.0f / M);

  // --- layer 2: agg(BN+ReLU(bufA)) -> bufB ; bufB @ W2 -> bufA ; stats ---
  k_aggregate<<<(G * L * (C1 / 4)) / 256, blk, 0, stream>>>(
      bufA, bufB, csrStart, csrSrc, csrW, self_w, bnA, bnB, C1 / 4, L, 1);
  k_gemm_wmma<128, 128><<<dim3(M / 128, C2 / 16), blk, 0, stream>>>(bufB, W2, b2, bufA);
  k_stats_partial<<<M / 256, blk, 0, stream>>>(bufA, psum, psq, C2);
  k_stats_final<<<1, 128, 0, stream>>>(psum, psq, g2, be2, bnA, bnB, C2, M / 256, 1.0f / M);

  // --- layer 3: agg(BN+ReLU(bufA)) -> bufB ; bufB @ W3 -> bufA ; stats ---
  k_aggregate<<<(G * L * (C2 / 4)) / 256, blk, 0, stream>>>(
      bufA, bufB, csrStart, csrSrc, csrW, self_w, bnA, bnB, C2 / 4, L, 1);
  k_gemm_wmma<128, 64><<<dim3(M / 128, C3 / 16), blk, 0, stream>>>(bufB, W3, b3, bufA);
  k_stats_partial<<<M / 256, blk, 0, stream>>>(bufA, psum, psq, C3);
  k_stats_final<<<1, 128, 0, stream>>>(psum, psq, g3, be3, bnA, bnB, C3, M / 256, 1.0f / M);

  // --- output: BN+ReLU + transpose to [G, C3, L] ---
  k_output<<<(G * C3 * L) / 256, blk, 0, stream>>>(bufA, bnA, bnB, (float*)d_out, C3, L);
}